// GeneratorArchC_55448027791363
// MI455X (gfx1250) — compile-verified
//
#include <hip/hip_runtime.h>
#include <hip/hip_bf16.h>

typedef __attribute__((ext_vector_type(2))) float v2f;
typedef __attribute__((ext_vector_type(8))) float v8f;

#define NQ 6
#define NL 2

// ---------------------------------------------------------------------------
// Kernel 1: build signed columns of the fixed circuit unitary.
// Lane j (j<16) simulates U * e_{4j}  (basis state with b4=b5=0, j = b0b1b2b3),
// where U = [layers of CNOT-ring + RY] * RY(anc0 on q4) * RY(anc1 on q5).
// All gates are real; state is 64 floats fully held in registers (unrolled).
// Output (K-pair interleaved for float2 fragment loads):
//   W[(row>>1)*128 + n*2 + (row&1)] = sign(j) * U[n][4j],
// where row = par*8 + kk, kk = rank of j within its popcount-parity list.
// ---------------------------------------------------------------------------
__global__ void build_u_cols(const float* __restrict__ weights,
                             const float* __restrict__ anc,
                             float* __restrict__ W) {
    int j = threadIdx.x;
    if (j >= 16) return;

    float st[64];
#pragma unroll
    for (int i = 0; i < 64; ++i) st[i] = (i == 4 * j) ? 1.0f : 0.0f;

    auto RY = [&](int q, float theta) {
        float c = __cosf(0.5f * theta);
        float s = __sinf(0.5f * theta);
        int m = 1 << (5 - q);
#pragma unroll
        for (int i = 0; i < 64; ++i) {
            if ((i & m) == 0) {
                float v0 = st[i], v1 = st[i | m];
                st[i]     = c * v0 - s * v1;
                st[i | m] = s * v0 + c * v1;
            }
        }
    };
    auto CN = [&](int c, int t) {
        int mc = 1 << (5 - c), mt = 1 << (5 - t);
#pragma unroll
        for (int i = 0; i < 64; ++i) {
            if ((i & mc) && !(i & mt)) {
                float tmp = st[i]; st[i] = st[i | mt]; st[i | mt] = tmp;
            }
        }
    };

    RY(4, anc[0]);
    RY(5, anc[1]);
#pragma unroll
    for (int l = 0; l < NL; ++l) {
#pragma unroll
        for (int w = 0; w < NQ; ++w) CN(w, (w + 1) % NQ);
#pragma unroll
        for (int w = 0; w < NQ; ++w) RY(w, weights[l * NQ + w]);
    }

    // sign of (-i)^pop folded into the column:
    // pop%4==0 -> +Re, 1 -> -Im, 2 -> -Re, 3 -> +Im
    int pop = __popc(j);
    float sgn = (((pop + 1) >> 1) & 1) ? -1.0f : 1.0f;
    int par = pop & 1;
    int kk = 0;
    for (int jj = 0; jj < j; ++jj)
        if ((__popc(jj) & 1) == par) ++kk;

    int row = par * 8 + kk;
    float* dst = W + (row >> 1) * 128 + (row & 1);
#pragma unroll
    for (int n = 0; n < 64; ++n) dst[n * 2] = sgn * st[n];
}

// ---------------------------------------------------------------------------
// Kernel 2: batched sparse-K GEMM via V_WMMA_F32_16X16X4_F32.
//   Re(B x 64) = A_even(B x 8) @ W_even(8 x 64)   (even-popcount j's)
//   Im(B x 64) = A_odd (B x 8) @ W_odd (8 x 64)   (odd-popcount  j's)
//   out[m][w]  = sum_n sign_w(n) * (Re^2 + Im^2)
// Each wave handles 16-sample tiles: 16 WMMAs per tile.
// ---------------------------------------------------------------------------
__global__ void __launch_bounds__(256, 1)
qsim_wmma(const float* __restrict__ x,
          const float* __restrict__ W,
          float* __restrict__ out, int nTiles) {
    const int lane = threadIdx.x & 31;
    const int wid  = threadIdx.x >> 5;
    const int wavesPerBlock = blockDim.x >> 5;
    const int gwave  = blockIdx.x * wavesPerBlock + wid;
    const int nWaves = gridDim.x * wavesPerBlock;
    const int h   = lane >> 4;   // lane half: selects K pair {0,1} vs {2,3}
    const int n16 = lane & 15;
    const bool hi = (h != 0);

    // B fragments (4x16 f32): VGPR0 = row k0 (lanes0-15) / k0+2 (lanes16-31),
    // VGPR1 = k0+1 / k0+3; N across lanes. One float2 load per fragment.
    v2f bf[2][2][4];
#pragma unroll
    for (int p = 0; p < 2; ++p)
#pragma unroll
        for (int q = 0; q < 2; ++q) {
            int r = p * 4 + 2 * q + h;          // (row0 >> 1), row0 = p*8+4q+2h
#pragma unroll
            for (int t = 0; t < 4; ++t)
                bf[p][q][t] = *(const v2f*)(W + r * 128 + (16 * t + n16) * 2);
        }

    for (int tile = gwave; tile < nTiles; tile += nWaves) {
        // A-matrix M index for this lane (both halves cover M = lane&15)
        int m = tile * 16 + n16;
        float4 xv = *(const float4*)(x + m * 4);

        float s0, c0, s1, c1, s2, c2, s3, c3;
        __sincosf(xv.x * 0.5f, &s0, &c0);
        __sincosf(xv.y * 0.5f, &s1, &c1);
        __sincosf(xv.z * 0.5f, &s2, &c2);
        __sincosf(xv.w * 0.5f, &s3, &c3);

        // prod_j = prod_w (bit(3-w) of j ? sin_w : cos_w), j = b0b1b2b3
        float pa0 = c0 * c1, pa1 = c0 * s1, pa2 = s0 * c1, pa3 = s0 * s1;
        float pb0 = c2 * c3, pb1 = c2 * s3, pb2 = s2 * c3, pb3 = s2 * s3;
        float p0  = pa0 * pb0, p1  = pa0 * pb1, p2  = pa0 * pb2, p3  = pa0 * pb3;
        float p4  = pa1 * pb0, p5  = pa1 * pb1, p6  = pa1 * pb2, p7  = pa1 * pb3;
        float p8  = pa2 * pb0, p9  = pa2 * pb1, p10 = pa2 * pb2, p11 = pa2 * pb3;
        float p12 = pa3 * pb0, p13 = pa3 * pb1, p14 = pa3 * pb2, p15 = pa3 * pb3;

        // A fragments (16x4): lanes0-15 K={4q,4q+1}, lanes16-31 K={4q+2,4q+3}.
        // K order = parity lists: even {0,3,5,6, 9,10,12,15}, odd {1,2,4,7, 8,11,13,14}
        v2f ae0, ae1, ao0, ao1;
        ae0.x = hi ? p5  : p0;   ae0.y = hi ? p6  : p3;
        ae1.x = hi ? p12 : p9;   ae1.y = hi ? p15 : p10;
        ao0.x = hi ? p4  : p1;   ao0.y = hi ? p7  : p2;
        ao1.x = hi ? p13 : p8;   ao1.y = hi ? p14 : p11;

        // Sign accumulators:
        //   S0: signed by tile bit1 (n bit5, <Z0>), S1: by tile bit0 (n bit4, <Z1>)
        //   Sa: plain sum; lane-signs for <Z2>/<Z3> applied at reduction time.
        float S0[8], S1[8], Sa[8];
#pragma unroll
        for (int v = 0; v < 8; ++v) { S0[v] = 0.0f; S1[v] = 0.0f; Sa[v] = 0.0f; }

#pragma unroll
        for (int t = 0; t < 4; ++t) {
            v8f re = {0, 0, 0, 0, 0, 0, 0, 0};
            v8f im = {0, 0, 0, 0, 0, 0, 0, 0};
            re = __builtin_amdgcn_wmma_f32_16x16x4_f32(
                    false, ae0, false, bf[0][0][t], (short)0, re, false, false);
            re = __builtin_amdgcn_wmma_f32_16x16x4_f32(
                    false, ae1, false, bf[0][1][t], (short)0, re, false, false);
            im = __builtin_amdgcn_wmma_f32_16x16x4_f32(
                    false, ao0, false, bf[1][0][t], (short)0, im, false, false);
            im = __builtin_amdgcn_wmma_f32_16x16x4_f32(
                    false, ao1, false, bf[1][1][t], (short)0, im, false, false);
#pragma unroll
            for (int v = 0; v < 8; ++v) {
                float pv = __builtin_fmaf(re[v], re[v], im[v] * im[v]);
                Sa[v] += pv;
                if (t & 2) S0[v] -= pv; else S0[v] += pv;
                if (t & 1) S1[v] -= pv; else S1[v] += pv;
            }
        }

        // Reduce over N (16-lane halves; xor masks 1..8 stay within a half).
        // <Z2>/<Z3> share Sa: lane-sign bits (3 and 2) are constant within
        // xor-1/xor-2 groups, so reduce those two steps once, then sign+finish.
        float o0[8], o1[8], o2[8], o3[8];
#pragma unroll
        for (int v = 0; v < 8; ++v) {
            float r0 = S0[v];
            float r1 = S1[v];
            float a  = Sa[v];
            r0 += __shfl_xor(r0, 1, 32);
            r1 += __shfl_xor(r1, 1, 32);
            a  += __shfl_xor(a, 1, 32);
            r0 += __shfl_xor(r0, 2, 32);
            r1 += __shfl_xor(r1, 2, 32);
            a  += __shfl_xor(a, 2, 32);
            float r2 = (lane & 8) ? -a : a;   // n bit3 -> <Z2>
            float r3 = (lane & 4) ? -a : a;   // n bit2 -> <Z3>
            r0 += __shfl_xor(r0, 4, 32);
            r1 += __shfl_xor(r1, 4, 32);
            r2 += __shfl_xor(r2, 4, 32);
            r3 += __shfl_xor(r3, 4, 32);
            r0 += __shfl_xor(r0, 8, 32);
            r1 += __shfl_xor(r1, 8, 32);
            r2 += __shfl_xor(r2, 8, 32);
            r3 += __shfl_xor(r3, 8, 32);
            o0[v] = r0; o1[v] = r1; o2[v] = r2; o3[v] = r3;
        }

        if (n16 == 0) {
#pragma unroll
            for (int v = 0; v < 8; ++v) {
                int mrow = tile * 16 + v + 8 * h;   // D layout: M = v + 8*half
                float4 o = {o0[v], o1[v], o2[v], o3[v]};
                *(float4*)(out + mrow * 4) = o;
            }
        }
    }
}

extern "C" void kernel_launch(void* const* d_in, const int* in_sizes, int n_in,
                              void* d_out, int out_size, void* d_ws, size_t ws_size,
                              hipStream_t stream) {
    const float* x   = (const float*)d_in[0];  // (B, 4)
    const float* wts = (const float*)d_in[1];  // (2, 6)
    const float* anc = (const float*)d_in[2];  // (2,)
    float* out = (float*)d_out;                // (B, 4)
    float* W   = (float*)d_ws;                 // 16 x 64 floats = 4 KB (interleaved pairs)

    int Bn = in_sizes[0] / 4;
    int nTiles = Bn / 16;

    build_u_cols<<<dim3(1), dim3(32), 0, stream>>>(wts, anc, W);

    int blocks = (nTiles + 31) / 32;           // ~4 tiles per wave at 8 waves/block
    if (blocks < 1) blocks = 1;
    if (blocks > 256) blocks = 256;
    qsim_wmma<<<dim3(blocks), dim3(256), 0, stream>>>(x, W, out, nTiles);
}